// EntailmentConeLoss_54528904790726
// MI455X (gfx1250) — compile-verified
//
#include <hip/hip_runtime.h>
#include <hip/hip_bf16.h>
#include <math.h>
#include <stdint.h>

// ---------------- problem constants (from reference) ----------------
#define C_SZ   100000
#define D_SZ   256
#define P_SZ   65536            // groups: one positive pair + K negatives
#define K_NEG  4

#define BETA_F   0.1f
#define MARGIN_F 0.1f
#define EPS_F    1e-6f

// ---------------- launch geometry ----------------
#define BLOCK  128                    // 4 waves (wave32)
#define WPB    (BLOCK / 32)
#define GRID   2048
#define NWAVES (GRID * WPB)           // 8192 -> exactly 8 groups per wave

#define ROW_F   D_SZ                  // floats per row
#define ROW_B   (D_SZ * 4)            // bytes per row (1KB)
#define GROUP_ROWS 6                  // p, c, nc0..nc3
#define GBUF_F  (GROUP_ROWS * ROW_F)  // floats per group buffer
#define GBUF_B  (GROUP_ROWS * ROW_B)  // 6KB
#define WAVE_F  (2 * GBUF_F)          // double buffered, per wave (12KB)

// Async-copy one 1KB row (256 f32) into LDS. Each lane moves 2x16B.
// ASYNCcnt += 2.
__device__ __forceinline__ void async_copy_row_1k(const float* gsrc,
                                                  uint32_t lds_off, int lane) {
  uint64_t ga = (uint64_t)(uintptr_t)gsrc + (uint64_t)(lane * 16);
  uint32_t la = lds_off + (uint32_t)(lane * 16);
  asm volatile(
      "global_load_async_to_lds_b128 %0, %1, off\n\t"
      "global_load_async_to_lds_b128 %0, %1, off offset:512"
      :
      : "v"(la), "v"(ga)
      : "memory");
}

// Gather all 6 rows of group g into the LDS buffer at lds_off. ASYNCcnt += 12.
__device__ __forceinline__ void issue_group(int g,
                                            const float* __restrict__ prot,
                                            const int* __restrict__ pairs,
                                            const int* __restrict__ negc,
                                            uint32_t lds_off, int lane) {
  int2 pc = *(const int2*)(pairs + 2 * g);   // (parent, child)
  int4 nc = *(const int4*)(negc + 4 * g);    // 4 negative children
  async_copy_row_1k(prot + (size_t)pc.x * ROW_F, lds_off + 0 * ROW_B, lane);
  async_copy_row_1k(prot + (size_t)pc.y * ROW_F, lds_off + 1 * ROW_B, lane);
  async_copy_row_1k(prot + (size_t)nc.x * ROW_F, lds_off + 2 * ROW_B, lane);
  async_copy_row_1k(prot + (size_t)nc.y * ROW_F, lds_off + 3 * ROW_B, lane);
  async_copy_row_1k(prot + (size_t)nc.z * ROW_F, lds_off + 4 * ROW_B, lane);
  async_copy_row_1k(prot + (size_t)nc.w * ROW_F, lds_off + 5 * ROW_B, lane);
}

__device__ __forceinline__ float wave_sum32(float v) {
#pragma unroll
  for (int o = 16; o > 0; o >>= 1) v += __shfl_xor(v, o, 32);
  return v;
}

__global__ __launch_bounds__(BLOCK) void cone_loss_main(
    const float* __restrict__ prot, const int* __restrict__ pairs,
    const int* __restrict__ negc, float* __restrict__ partial /* [2*GRID] */) {
  __shared__ __align__(16) float smem[WPB * WAVE_F];   // 48 KB
  __shared__ float red_p[WPB];
  __shared__ float red_n[WPB];

  const int lane = threadIdx.x & 31;
  const int wv   = threadIdx.x >> 5;
  const int wgid = blockIdx.x * WPB + wv;

  float* wbase = &smem[wv * WAVE_F];
  // low 32 bits of a generic LDS pointer == workgroup-relative LDS byte offset
  const uint32_t wbase_off = (uint32_t)(uintptr_t)(void*)wbase;

  // per-lane energy-slot registers (32 pending energies per wave)
  float s_cc = 0.0f, s_dd = 0.0f, s_pp = 0.0f, s_fl = 0.0f;
  int   cnt  = 0;
  float accp = 0.0f, accn = 0.0f;   // per-lane batched accumulators

  // vectorized finalize of up to 32 parked energies (one per lane)
  auto finalize = [&](int valid_count) {
    float num    = s_cc - s_pp - s_dd;
    float np     = sqrtf(s_pp);
    float den    = 2.0f * np * sqrtf(s_dd) + EPS_F;
    float cosang = fminf(fmaxf(num / den, -1.0f + EPS_F), 1.0f - EPS_F);
    float ang    = acosf(cosang);
    float apert  = asinf(fminf(fmaxf(BETA_F / (np + EPS_F), 0.0f),
                               1.0f - EPS_F));
    float energy = fmaxf(ang - apert, 0.0f);
    bool  valid  = lane < valid_count;
    bool  pos    = s_fl > 0.5f;
    accp += (valid && pos)  ? energy : 0.0f;
    accn += (valid && !pos) ? fmaxf(MARGIN_F - energy, 0.0f) : 0.0f;
  };

  int g   = wgid;
  int buf = 0;
  if (g < P_SZ) {
    issue_group(g, prot, pairs, negc, wbase_off + (uint32_t)buf * GBUF_B, lane);
  }

  for (; g < P_SZ; g += NWAVES) {
    const int nxt = g + NWAVES;
    if (nxt < P_SZ) {
      issue_group(nxt, prot, pairs, negc,
                  wbase_off + (uint32_t)(buf ^ 1) * GBUF_B, lane);
      asm volatile("s_wait_asynccnt 12" ::: "memory");
    } else {
      asm volatile("s_wait_asynccnt 0" ::: "memory");
    }

    const float* B = wbase + buf * GBUF_F;

    // parent row: lane holds elems [4L..4L+3] and [128+4L..128+4L+3]
    const float4* p4 = (const float4*)B;
    float4 pa = p4[lane];
    float4 pb = p4[lane + 32];
    float px[8] = {pa.x, pa.y, pa.z, pa.w, pb.x, pb.y, pb.z, pb.w};

    float ppart = 0.0f;
#pragma unroll
    for (int i = 0; i < 8; ++i) ppart = fmaf(px[i], px[i], ppart);
    float pp = wave_sum32(ppart);     // shared by all 5 energies of the group

#pragma unroll
    for (int r = 0; r < 1 + K_NEG; ++r) {
      const float4* c4 = (const float4*)(B + (1 + r) * ROW_F);
      float4 ca = c4[lane];
      float4 cb = c4[lane + 32];
      float cx[8] = {ca.x, ca.y, ca.z, ca.w, cb.x, cb.y, cb.z, cb.w};

      float ccp = 0.0f, ddp = 0.0f;
#pragma unroll
      for (int i = 0; i < 8; ++i) {
        float d = cx[i] - px[i];
        ccp = fmaf(cx[i], cx[i], ccp);
        ddp = fmaf(d, d, ddp);
      }
      float cc = wave_sum32(ccp);
      float dd = wave_sum32(ddp);

      // park this energy's reduced sums in slot `cnt` (lane == cnt)
      bool mine = (lane == cnt);
      s_cc = mine ? cc : s_cc;
      s_dd = mine ? dd : s_dd;
      s_pp = mine ? pp : s_pp;
      s_fl = mine ? ((r == 0) ? 1.0f : 0.0f) : s_fl;
      if (++cnt == 32) { finalize(32); cnt = 0; }
    }

    buf ^= 1;
  }

  if (cnt > 0) finalize(cnt);   // residual slots

  float tp = wave_sum32(accp);
  float tn = wave_sum32(accn);
  if (lane == 0) { red_p[wv] = tp; red_n[wv] = tn; }
  __syncthreads();
  if (threadIdx.x == 0) {
    float sp = 0.0f, sn = 0.0f;
#pragma unroll
    for (int i = 0; i < WPB; ++i) { sp += red_p[i]; sn += red_n[i]; }
    partial[blockIdx.x]        = sp;
    partial[GRID + blockIdx.x] = sn;
  }
}

#define FBLOCK 256
__global__ __launch_bounds__(FBLOCK) void cone_loss_final(
    const float* __restrict__ partial, float* __restrict__ out) {
  __shared__ float sp[FBLOCK];
  __shared__ float sn[FBLOCK];
  float ap = 0.0f, an = 0.0f;
  for (int i = threadIdx.x; i < GRID; i += FBLOCK) {
    ap += partial[i];
    an += partial[GRID + i];
  }
  sp[threadIdx.x] = ap;
  sn[threadIdx.x] = an;
  __syncthreads();
  for (int s = FBLOCK / 2; s > 0; s >>= 1) {
    if (threadIdx.x < s) {
      sp[threadIdx.x] += sp[threadIdx.x + s];
      sn[threadIdx.x] += sn[threadIdx.x + s];
    }
    __syncthreads();
  }
  if (threadIdx.x == 0) {
    float avg_pos = sp[0] / (float)P_SZ;
    float avg_neg = sn[0] / (float)(P_SZ * K_NEG);
    out[0] = 0.5f * (avg_pos + avg_neg);   // LOSS_WEIGHT = 1.0
  }
}

extern "C" void kernel_launch(void* const* d_in, const int* in_sizes, int n_in,
                              void* d_out, int out_size, void* d_ws,
                              size_t ws_size, hipStream_t stream) {
  const float* prot  = (const float*)d_in[0];  // (C, D) f32
  const int*   pairs = (const int*)d_in[1];    // (P, 2) i32
  const int*   negc  = (const int*)d_in[2];    // (P*K,) i32
  float* partial = (float*)d_ws;               // needs 2*GRID floats (16 KB)
  float* out     = (float*)d_out;

  hipLaunchKernelGGL(cone_loss_main, dim3(GRID), dim3(BLOCK), 0, stream,
                     prot, pairs, negc, partial);
  hipLaunchKernelGGL(cone_loss_final, dim3(1), dim3(FBLOCK), 0, stream,
                     partial, out);
}